// EdgeClassify_5214090297364
// MI455X (gfx1250) — compile-verified
//
#include <hip/hip_runtime.h>
#include <hip/hip_bf16.h>

// Problem constants from the reference module.
#define B_  64
#define S_  2048
#define D_  1024
#define A_  13
#define NB_ 4

typedef __attribute__((ext_vector_type(2))) float v2f;
typedef __attribute__((ext_vector_type(8))) float v8f;

// One block per batch element b. 256 threads = 8 waves.
// Stage 1: red_tile[s][a] = dot(e_output[b,s,:], W1[a,:]) + b1[a]  (s,a < 13, padded to 16x16)
//          via V_WMMA_F32_16X16X4_F32, K split across the 8 waves, reduced in LDS.
// Stage 2: ga[i][n] = sum_s tile[s][i] * Wf[n, s]      (Wa)
//          gb[i][n] = sum_s tile[s][i] * Wf[n, 13+s]   (Wb)
// Stage 3: out[b,i,j,n] = (i!=j) ? ga[min][n] + gb[max][n] + bf[n] : 0
__global__ __launch_bounds__(256)
void edge_classify_kernel(const float* __restrict__ E,   // (B,S,D)
                          const float* __restrict__ W1,  // (A,D)
                          const float* __restrict__ b1,  // (A)
                          const float* __restrict__ Wf,  // (NB, 2A)
                          const float* __restrict__ bf,  // (NB)
                          float* __restrict__ out)       // (B,A,A,NB)
{
    __shared__ float s_partial[8 * 256];   // 8 waves x 16x16 f32 tile
    __shared__ float s_tile[16 * 16];      // reduced tile (+bias)
    __shared__ float s_g[2 * A_ * NB_];    // ga then gb

    const int b    = blockIdx.x;
    const int tid  = threadIdx.x;
    const int lane = tid & 31;
    const int wave = tid >> 5;
    const int half = lane >> 4;        // lane half selects K pair {0,1} vs {2,3}
    const int mn   = lane & 15;

    // A-matrix row: sequence position m (rows 13..15 read valid memory, values unused)
    const float* __restrict__ Arow = E + ((size_t)b * S_ + mn) * D_;
    // B-matrix col: atom index (clamp 13..15 into bounds; those columns are unused)
    const int nc = (mn < A_) ? mn : (A_ - 1);
    const float* __restrict__ Brow = W1 + (size_t)nc * D_;

    v8f c = {};
    const int kbase = wave * (D_ / 8);   // 128 K per wave
    #pragma unroll 4
    for (int k0 = 0; k0 < D_ / 8; k0 += 4) {
        const int ka = kbase + k0 + 2 * half;
        v2f av; av[0] = Arow[ka]; av[1] = Arow[ka + 1];
        v2f bv; bv[0] = Brow[ka]; bv[1] = Brow[ka + 1];
        // 8 args: (neg_a, A, neg_b, B, c_mod, C, reuse_a, reuse_b)
        c = __builtin_amdgcn_wmma_f32_16x16x4_f32(false, av, false, bv,
                                                  (short)0, c, false, false);
    }

    // Spill this wave's accumulator tile to LDS.
    #pragma unroll
    for (int r = 0; r < 8; ++r)
        s_partial[wave * 256 + r * 32 + lane] = c[r];
    __syncthreads();

    // Cross-wave reduction: thread `tid` owns one of the 256 tile slots.
    {
        float sum = 0.0f;
        #pragma unroll
        for (int w = 0; w < 8; ++w)
            sum += s_partial[w * 256 + tid];
        const int r = tid >> 5;
        const int l = tid & 31;
        const int M = r + 8 * ((l >> 4) & 1);  // sequence s
        const int N = l & 15;                  // atom col i
        s_tile[M * 16 + N] = sum + (N < A_ ? b1[N] : 0.0f);
    }
    __syncthreads();

    // Stage 2: 104 threads compute ga (kind=0) / gb (kind=1), each a 13-dot.
    if (tid < 2 * A_ * NB_) {
        const int kind = tid / (A_ * NB_);
        const int rem  = tid % (A_ * NB_);
        const int i    = rem >> 2;
        const int n    = rem & 3;
        const float* __restrict__ wrow = Wf + n * (2 * A_) + kind * A_;
        float acc = 0.0f;
        #pragma unroll
        for (int s = 0; s < A_; ++s)
            acc += s_tile[s * 16 + i] * wrow[s];
        s_g[kind * (A_ * NB_) + i * NB_ + n] = acc;
    }
    __syncthreads();

    // Stage 3: assemble out[b, i, j, n]; diagonal zeroed.
    float* __restrict__ outb = out + (size_t)b * (A_ * A_ * NB_);
    for (int e = tid; e < A_ * A_ * NB_; e += 256) {
        const int n  = e & 3;
        const int ij = e >> 2;
        const int j  = ij % A_;
        const int i  = ij / A_;
        float v = 0.0f;
        if (i != j) {
            const int lo = (i < j) ? i : j;
            const int hi = (i < j) ? j : i;
            v = s_g[lo * NB_ + n] + s_g[A_ * NB_ + hi * NB_ + n] + bf[n];
        }
        outb[e] = v;
    }
}

extern "C" void kernel_launch(void* const* d_in, const int* in_sizes, int n_in,
                              void* d_out, int out_size, void* d_ws, size_t ws_size,
                              hipStream_t stream) {
    (void)in_sizes; (void)n_in; (void)d_ws; (void)ws_size; (void)out_size;
    const float* E  = (const float*)d_in[0];   // e_output (B,S,D)
    const float* W1 = (const float*)d_in[1];   // (A,D)
    const float* b1 = (const float*)d_in[2];   // (A)
    const float* Wf = (const float*)d_in[3];   // (NB,2A)
    const float* bf = (const float*)d_in[4];   // (NB)
    float* out = (float*)d_out;                // (B,A,A,NB)

    edge_classify_kernel<<<dim3(B_), dim3(256), 0, stream>>>(E, W1, b1, Wf, bf, out);
}